// KGE_74113955660380
// MI455X (gfx1250) — compile-verified
//
#include <hip/hip_runtime.h>
#include <math.h>

typedef __attribute__((ext_vector_type(2))) float v2f;
typedef __attribute__((ext_vector_type(8))) float v8f;

#define NP 8000
#define NV 3000
#define NC 1500
#define NA 6000
#define CLSN 1500
#define DIM 128
#define NEDGE 150000
#define NTOT 18500

// ---------------------------------------------------------------------------
// WMMA helper: D = A(16x4 f32) x B(4x16 f32) + C(16x16 f32)
// ---------------------------------------------------------------------------
__device__ __forceinline__ v8f wmma_f32x4(v2f a, v2f b, v8f c) {
  return __builtin_amdgcn_wmma_f32_16x16x4_f32(false, a, false, b, (short)0, c,
                                               false, false);
}

// ---------------------------------------------------------------------------
// C[M x N] = X[M x K] @ W[N x K]^T + bias[N]
// one wave -> one 16x16 tile; block = 4 waves (4 row tiles).
// OOB rows/cols use clamped (valid, finite) addresses: WMMA contributions are
// separable, so garbage only lands in never-stored OOB outputs -> no masking
// VALU in the inner loop.
// ---------------------------------------------------------------------------
__global__ __launch_bounds__(128) void gemm_xwT_wmma(
    const float* __restrict__ X, const float* __restrict__ W,
    const float* __restrict__ bias, float* __restrict__ C, int M, int N,
    int K) {
  const int wave = threadIdx.x >> 5;
  const int lane = threadIdx.x & 31;
  const int half = lane >> 4;
  const int l = lane & 15;
  const int tileM = blockIdx.y * 4 + wave;
  const int tileN = blockIdx.x;

  const int row = tileM * 16 + l;
  const int col = tileN * 16 + l;
  const int arow = (row < M) ? row : (M - 1);
  const int wrow = (col < N) ? col : (N - 1);

  const float* __restrict__ xp = X + (size_t)arow * K + 2 * half;
  const float* __restrict__ wp = W + (size_t)wrow * K + 2 * half;

  v8f acc = {0.f, 0.f, 0.f, 0.f, 0.f, 0.f, 0.f, 0.f};
#pragma unroll 4
  for (int k = 0; k < K; k += 4) {
    const v2f a = *(const v2f*)(xp + k);
    const v2f b = *(const v2f*)(wp + k);
    acc = wmma_f32x4(a, b, acc);
  }

  const float badd = (bias != nullptr && col < N) ? bias[col] : 0.f;
#pragma unroll
  for (int r = 0; r < 8; ++r) {
    const int orow = tileM * 16 + r + 8 * half;
    if (orow < M && col < N) C[(size_t)orow * N + col] = acc[r] + badd;
  }
}

// ---------------------------------------------------------------------------
// C[M x 128] = A[M x K] @ (tcp[k] * Y[K x 128])
// block = 128 threads (4 waves) -> 64 rows x 128 cols of C.
// Y tiles (32 x 128, tcp-scaled) staged through LDS (reused operand, keep in
// L2); adjacency A streamed once from HBM with non-temporal loads + prefetch.
// ---------------------------------------------------------------------------
__global__ __launch_bounds__(128) void gemm_adj_wmma(
    const float* __restrict__ A, const float* __restrict__ Y,
    const float* __restrict__ tcp, float* __restrict__ C, int M, int K) {
  __shared__ float ldsY[32 * DIM];

  const int tid = threadIdx.x;
  const int wave = tid >> 5;
  const int lane = tid & 31;
  const int half = lane >> 4;
  const int l = lane & 15;
  const int tileM = blockIdx.x * 4 + wave;

  const int row = tileM * 16 + l;
  const int arow = (row < M) ? row : (M - 1);
  const float* __restrict__ ap = A + (size_t)arow * K + 2 * half;

  v8f acc[8] = {};

  for (int k0 = 0; k0 < K; k0 += 32) {
    __syncthreads();
    // cooperative stage of tcp-scaled Y[k0..k0+32) x 128 into LDS
    for (int idx = tid; idx < 32 * DIM; idx += 128) {
      const int kr = idx >> 7;
      const int c = idx & (DIM - 1);
      const int kg = k0 + kr;
      float v = 0.f;
      if (kg < K) v = tcp[kg] * Y[(size_t)kg * DIM + c];
      ldsY[idx] = v;
    }
    __syncthreads();
    // prefetch the next adjacency chunk while we compute on this one
    __builtin_prefetch(ap + k0 + 32, 0, 1);

#pragma unroll
    for (int g = 0; g < 8; ++g) {
      const int kb = k0 + 4 * g;
      if (kb < K) {  // uniform: K % 4 == 0 so in-range groups are full
        // adjacency is single-use: non-temporal so it does not evict the
        // L2-resident Y / weight operands
        const v2f a = __builtin_nontemporal_load((const v2f*)(ap + kb));
        const int krow = 4 * g + 2 * half;
#pragma unroll
        for (int ct = 0; ct < 8; ++ct) {
          v2f b;
          b.x = ldsY[krow * DIM + ct * 16 + l];
          b.y = ldsY[(krow + 1) * DIM + ct * 16 + l];
          acc[ct] = wmma_f32x4(a, b, acc[ct]);
        }
      }
    }
  }

#pragma unroll
  for (int ct = 0; ct < 8; ++ct) {
#pragma unroll
    for (int r = 0; r < 8; ++r) {
      const int orow = tileM * 16 + r + 8 * half;
      if (orow < M) C[(size_t)orow * DIM + ct * 16 + l] = acc[ct][r];
    }
  }
}

// ---------------------------------------------------------------------------
// block reductions
// ---------------------------------------------------------------------------
__device__ __forceinline__ float blk_max(float v, float* red) {
  const int t = threadIdx.x;
  red[t] = v;
  __syncthreads();
  for (int s = blockDim.x >> 1; s > 0; s >>= 1) {
    if (t < s) red[t] = fmaxf(red[t], red[t + s]);
    __syncthreads();
  }
  const float r = red[0];
  __syncthreads();
  return r;
}

__device__ __forceinline__ float blk_sum(float v, float* red) {
  const int t = threadIdx.x;
  red[t] = v;
  __syncthreads();
  for (int s = blockDim.x >> 1; s > 0; s >>= 1) {
    if (t < s) red[t] += red[t + s];
    __syncthreads();
  }
  const float r = red[0];
  __syncthreads();
  return r;
}

// ---------------------------------------------------------------------------
// per-row softmax + soft cross-entropy + tcp = sum(pred*label)
// one block (256 threads) per row of [rows x CLSN]
// ---------------------------------------------------------------------------
__global__ __launch_bounds__(256) void softmax_ce_tcp(
    const float* __restrict__ logits, const float* __restrict__ labels, int C,
    float invN, float* __restrict__ tcp, float* __restrict__ loss_acc) {
  __shared__ float red[256];
  const int row = blockIdx.x;
  const float* x = logits + (size_t)row * C;
  const float* lb = labels + (size_t)row * C;

  float m = -3.4e38f;
  for (int c = threadIdx.x; c < C; c += blockDim.x) m = fmaxf(m, x[c]);
  m = blk_max(m, red);

  float se = 0.f, t1 = 0.f, sl = 0.f, t2 = 0.f;
  for (int c = threadIdx.x; c < C; c += blockDim.x) {
    const float xs = x[c] - m;
    const float e = expf(xs);
    const float lv = lb[c];
    se += e;
    t1 += lv * xs;
    sl += lv;
    t2 += lv * e;
  }
  se = blk_sum(se, red);
  t1 = blk_sum(t1, red);
  sl = blk_sum(sl, red);
  t2 = blk_sum(t2, red);

  if (threadIdx.x == 0) {
    tcp[row] = t2 / se;
    atomicAdd(loss_acc, -(t1 - logf(se) * sl) * invN);
  }
}

// ---------------------------------------------------------------------------
// GRU gate: out = (1-z)*tanh(in + r*hn) + z*h
// ---------------------------------------------------------------------------
__device__ __forceinline__ float sigmoidf(float x) {
  return 1.f / (1.f + expf(-x));
}

__global__ void gru_gate(const float* __restrict__ gi,
                         const float* __restrict__ gh,
                         const float* __restrict__ hin, float* __restrict__ out,
                         int total) {
  const int i = blockIdx.x * blockDim.x + threadIdx.x;
  if (i >= total) return;
  const int n = i >> 7;
  const int d = i & (DIM - 1);
  const size_t b = (size_t)n * (3 * DIM);
  const float r = sigmoidf(gi[b + d] + gh[b + d]);
  const float z = sigmoidf(gi[b + DIM + d] + gh[b + DIM + d]);
  const float nn = tanhf(gi[b + 2 * DIM + d] + r * gh[b + 2 * DIM + d]);
  out[i] = (1.f - z) * nn + z * hin[i];
}

// ---------------------------------------------------------------------------
// fuse: per-row 128-dot with a weight vector (wave32 per row)
// ---------------------------------------------------------------------------
__global__ void row_dot(const float* __restrict__ X, const float* __restrict__ w,
                        const float* __restrict__ b, float* __restrict__ out,
                        int N) {
  const int gw = (blockIdx.x * blockDim.x + threadIdx.x) >> 5;
  const int lane = threadIdx.x & 31;
  if (gw >= N) return;
  const float* x = X + (size_t)gw * DIM;
  float s = 0.f;
  for (int d = lane; d < DIM; d += 32) s += x[d] * w[d];
  s += __shfl_down(s, 16);
  s += __shfl_down(s, 8);
  s += __shfl_down(s, 4);
  s += __shfl_down(s, 2);
  s += __shfl_down(s, 1);
  if (lane == 0) out[gw] = s + b[0];
}

// softmax over the N dimension of relu(s): s[i] <- exp(relu(s[i])-m)/sum
__global__ __launch_bounds__(1024) void softmax_relu_vec(float* __restrict__ s,
                                                         int N) {
  __shared__ float red[1024];
  float m = -3.4e38f;
  for (int i = threadIdx.x; i < N; i += blockDim.x)
    m = fmaxf(m, fmaxf(s[i], 0.f));
  m = blk_max(m, red);
  float se = 0.f;
  for (int i = threadIdx.x; i < N; i += blockDim.x)
    se += expf(fmaxf(s[i], 0.f) - m);
  se = blk_sum(se, red);
  for (int i = threadIdx.x; i < N; i += blockDim.x)
    s[i] = expf(fmaxf(s[i], 0.f) - m) / se;
}

__global__ void fuse_combine(const float* __restrict__ e0,
                             const float* __restrict__ e1,
                             const float* __restrict__ a0,
                             const float* __restrict__ a1,
                             float* __restrict__ out, int total) {
  const int i = blockIdx.x * blockDim.x + threadIdx.x;
  if (i >= total) return;
  const int n = i >> 7;
  out[i] = fmaxf(a0[n] * e0[i] + a1[n] * e1[i], 0.f);
}

// ---------------------------------------------------------------------------
// edge losses: wave32 per edge, shuffle reduce, one atomic per block
// ---------------------------------------------------------------------------
__device__ __forceinline__ float neg_logsigmoid(float x) {
  // -logsigmoid(x) = -(min(x,0) - log1p(exp(-|x|)))
  return -(fminf(x, 0.f) - log1pf(expf(-fabsf(x))));
}

__global__ __launch_bounds__(256) void edge_loss_s(
    const int* __restrict__ us, const float* __restrict__ ent,
    float* __restrict__ acc, int E) {
  __shared__ float red[8];
  const int gw = (blockIdx.x * blockDim.x + threadIdx.x) >> 5;
  const int lane = threadIdx.x & 31;
  const int w = threadIdx.x >> 5;
  float c = 0.f;
  if (gw < E) {
    const int i = us[gw * 3 + 0];
    const int j = us[gw * 3 + 1];
    const float lab = (float)us[gw * 3 + 2];
    const float* pa = ent + (size_t)i * DIM;
    const float* pb = ent + (size_t)j * DIM;
    float s = 0.f;
    for (int d = lane; d < DIM; d += 32) s += pa[d] * pb[d];
    s += __shfl_down(s, 16);
    s += __shfl_down(s, 8);
    s += __shfl_down(s, 4);
    s += __shfl_down(s, 2);
    s += __shfl_down(s, 1);
    if (lane == 0) c = neg_logsigmoid(lab * s);
  }
  if (lane == 0) red[w] = c;
  __syncthreads();
  if (threadIdx.x == 0) {
    float t = 0.f;
    for (int q = 0; q < 8; ++q) t += red[q];
    atomicAdd(acc, t);
  }
}

__global__ __launch_bounds__(256) void edge_loss_d(
    const int* __restrict__ ud, const float* __restrict__ ent,
    const float* __restrict__ rel, float* __restrict__ acc, int E) {
  __shared__ float red[8];
  const int gw = (blockIdx.x * blockDim.x + threadIdx.x) >> 5;
  const int lane = threadIdx.x & 31;
  const int w = threadIdx.x >> 5;
  float c = 0.f;
  if (gw < E) {
    const int i = ud[gw * 4 + 0];
    const int j = ud[gw * 4 + 1];
    const float lab = (float)ud[gw * 4 + 2];
    const int r = ud[gw * 4 + 3];
    const float* pa = ent + (size_t)i * DIM;
    const float* pb = ent + (size_t)j * DIM;
    const float* pr = rel + (size_t)r * DIM;
    float s = 0.f;
    for (int d = lane; d < DIM; d += 32) s += pr[d] * tanhf(pa[d] + pb[d]);
    s += __shfl_down(s, 16);
    s += __shfl_down(s, 8);
    s += __shfl_down(s, 4);
    s += __shfl_down(s, 2);
    s += __shfl_down(s, 1);
    if (lane == 0) c = neg_logsigmoid(lab * s);
  }
  if (lane == 0) red[w] = c;
  __syncthreads();
  if (threadIdx.x == 0) {
    float t = 0.f;
    for (int q = 0; q < 8; ++q) t += red[q];
    atomicAdd(acc, t);
  }
}

// ---------------------------------------------------------------------------
// misc
// ---------------------------------------------------------------------------
__global__ void zero_scalars(float* p) {
  if (threadIdx.x < 16) p[threadIdx.x] = 0.f;
}

__global__ void finalize_out(const float* __restrict__ acc,
                             const float* __restrict__ rel,
                             float* __restrict__ out) {
  const int i = blockIdx.x * blockDim.x + threadIdx.x;
  if (i == 0) out[0] = acc[0];
  if (i < 12 * DIM) out[1 + (size_t)NTOT * DIM + i] = rel[i];
}

// ---------------------------------------------------------------------------
// launch
// ---------------------------------------------------------------------------
extern "C" void kernel_launch(void* const* d_in, const int* in_sizes, int n_in,
                              void* d_out, int out_size, void* d_ws,
                              size_t ws_size, hipStream_t stream) {
  (void)in_sizes; (void)n_in; (void)out_size; (void)ws_size;

  const float* feat_P = (const float*)d_in[0];
  const float* feat_V = (const float*)d_in[1];
  const float* feat_C = (const float*)d_in[2];
  const float* feat_A = (const float*)d_in[3];
  const float* W1 = (const float*)d_in[4];
  const float* b1 = (const float*)d_in[5];
  const float* W2 = (const float*)d_in[6];
  const float* b2 = (const float*)d_in[7];
  const float* W3 = (const float*)d_in[8];
  const float* b3 = (const float*)d_in[9];
  const float* L00W = (const float*)d_in[10];
  const float* L00b = (const float*)d_in[11];
  const float* L01W = (const float*)d_in[12];
  const float* L01b = (const float*)d_in[13];
  const float* L10W = (const float*)d_in[14];
  const float* L10b = (const float*)d_in[15];
  const float* L11W = (const float*)d_in[16];
  const float* L11b = (const float*)d_in[17];
  const float* L12W = (const float*)d_in[18];
  const float* L12b = (const float*)d_in[19];
  const float* g0_Wih = (const float*)d_in[20];
  const float* g0_Whh = (const float*)d_in[21];
  const float* g0_bih = (const float*)d_in[22];
  const float* g0_bhh = (const float*)d_in[23];
  const float* g1_Wih = (const float*)d_in[24];
  const float* g1_Whh = (const float*)d_in[25];
  const float* g1_bih = (const float*)d_in[26];
  const float* g1_bhh = (const float*)d_in[27];
  const float* rel_emb = (const float*)d_in[28];
  const float* adj00 = (const float*)d_in[29];
  const float* adj01 = (const float*)d_in[30];
  const float* adj10 = (const float*)d_in[31];
  const float* adj11 = (const float*)d_in[32];
  const float* adj12 = (const float*)d_in[33];
  const float* clsA = (const float*)d_in[34];
  const float* clsP = (const float*)d_in[35];
  const float* clsV = (const float*)d_in[36];
  const int* u_s = (const int*)d_in[37];
  const int* u_d = (const int*)d_in[38];

  // workspace layout (floats)
  float* ws = (float*)d_ws;
  float* acc = ws;  // 16 scalars
  float* selfP = ws + 16;
  float* selfV = selfP + (size_t)NP * DIM;
  float* selfC = selfV + (size_t)NV * DIM;
  float* selfA = selfC + (size_t)NC * DIM;
  float* stP0 = selfA + (size_t)NA * DIM;
  float* stA0 = stP0 + (size_t)NP * DIM;
  float* stP1 = stA0 + (size_t)NA * DIM;
  float* stV1 = stP1 + (size_t)NP * DIM;
  float* stC1 = stV1 + (size_t)NV * DIM;
  float* hbuf = stC1 + (size_t)NC * DIM;
  float* gi = hbuf + (size_t)NP * DIM;
  float* gh = gi + (size_t)NP * 3 * DIM;
  float* logits = gh + (size_t)NP * 3 * DIM;
  float* tcp = logits + (size_t)NP * CLSN;
  float* s0 = tcp + NP;
  float* s1 = s0 + NP;

  auto gemmXWT = [&](const float* X, const float* W, const float* bias,
                     float* C, int M, int N) {
    dim3 g((N + 15) / 16, (M + 63) / 64);
    gemm_xwT_wmma<<<g, 128, 0, stream>>>(X, W, bias, C, M, N, DIM);
  };
  auto gemmADJ = [&](const float* A, const float* Y, const float* t, float* C,
                     int M, int K) {
    gemm_adj_wmma<<<(M + 63) / 64, 128, 0, stream>>>(A, Y, t, C, M, K);
  };
  auto ce = [&](const float* lb, int M) {
    softmax_ce_tcp<<<M, 256, 0, stream>>>(logits, lb, CLSN, 1.f / (float)M,
                                          tcp, acc);
  };
  auto gru = [&](const float* x, const float* Wih, const float* bih,
                 const float* Whh, const float* bhh, float* out, int M) {
    gemmXWT(x, Wih, bih, gi, M, 3 * DIM);
    gemmXWT(hbuf, Whh, bhh, gh, M, 3 * DIM);
    const int tot = M * DIM;
    gru_gate<<<(tot + 255) / 256, 256, 0, stream>>>(gi, gh, hbuf, out, tot);
  };
  float* ent = (float*)d_out + 1;
  auto fuse = [&](const float* e0, const float* e1, float* out, int M) {
    const int wb = (M * 32 + 255) / 256;
    row_dot<<<wb, 256, 0, stream>>>(e0, W2, b2, s0, M);
    row_dot<<<wb, 256, 0, stream>>>(e1, W3, b3, s1, M);
    softmax_relu_vec<<<1, 1024, 0, stream>>>(s0, M);
    softmax_relu_vec<<<1, 1024, 0, stream>>>(s1, M);
    const int tot = M * DIM;
    fuse_combine<<<(tot + 255) / 256, 256, 0, stream>>>(e0, e1, s0, s1, out,
                                                        tot);
  };

  zero_scalars<<<1, 32, 0, stream>>>(acc);

  // self embeddings
  gemmXWT(feat_P, W1, b1, selfP, NP, DIM);
  gemmXWT(feat_V, W1, b1, selfV, NV, DIM);
  gemmXWT(feat_C, W1, b1, selfC, NC, DIM);
  gemmXWT(feat_A, W1, b1, selfA, NA, DIM);

  // stage: V -> P (gru0)
  gemmXWT(feat_V, L00W, L00b, logits, NV, CLSN);
  ce(clsV, NV);
  gemmADJ(adj00, feat_V, tcp, hbuf, NP, NV);
  gru(feat_P, g0_Wih, g0_bih, g0_Whh, g0_bhh, stP0, NP);

  // stage: P -> A (gru0)
  gemmXWT(stP0, L01W, L01b, logits, NP, CLSN);
  ce(clsP, NP);
  gemmADJ(adj01, stP0, tcp, hbuf, NA, NP);
  gru(feat_A, g0_Wih, g0_bih, g0_Whh, g0_bhh, stA0, NA);

  // stage: A -> P (gru1)
  gemmXWT(feat_A, L10W, L10b, logits, NA, CLSN);
  ce(clsA, NA);
  gemmADJ(adj10, feat_A, tcp, hbuf, NP, NA);
  gru(feat_P, g1_Wih, g1_bih, g1_Whh, g1_bhh, stP1, NP);

  // stage: P -> V (gru1)
  gemmXWT(stP1, L11W, L11b, logits, NP, CLSN);
  ce(clsP, NP);
  gemmADJ(adj11, stP1, tcp, hbuf, NV, NP);
  gru(feat_V, g1_Wih, g1_bih, g1_Whh, g1_bhh, stV1, NV);

  // stage: V -> C (gru1)
  gemmXWT(stV1, L12W, L12b, logits, NV, CLSN);
  ce(clsV, NV);
  gemmADJ(adj12, stV1, tcp, hbuf, NC, NV);
  gru(feat_C, g1_Wih, g1_bih, g1_Whh, g1_bhh, stC1, NC);

  // fuse into final_ent (d_out + 1), concat order [P, V, C, A]
  fuse(selfP, stP0, ent, NP);
  fuse(selfV, stV1, ent + (size_t)NP * DIM, NV);
  fuse(selfC, stC1, ent + (size_t)(NP + NV) * DIM, NC);
  fuse(selfA, stA0, ent + (size_t)(NP + NV + NC) * DIM, NA);

  // edge losses
  const int eb = (NEDGE * 32 + 255) / 256;
  edge_loss_s<<<eb, 256, 0, stream>>>(u_s, ent, acc, NEDGE);
  edge_loss_d<<<eb, 256, 0, stream>>>(u_d, ent, rel_emb, acc, NEDGE);

  // loss scalar + rel_emb passthrough
  finalize_out<<<(12 * DIM + 255) / 256, 256, 0, stream>>>(acc, rel_emb,
                                                           (float*)d_out);
}